// PointPrefilter_12816182411310
// MI455X (gfx1250) — compile-verified
//
#include <hip/hip_runtime.h>
#include <hip/hip_bf16.h>
#include <stdint.h>

// ---------------------------------------------------------------------------
// Farthest Point Sampling (persistent cooperative kernel) + async gather.
//
// ws layout:
//   [0      .. 8192)   : u64 best[1024]   (packed {dist_bits:32, ~index:32})
//   [8192   .. 8196)   : u32 barrier counter (monotonic)
//   [8448   .. 12544)  : i32 sel[1024]    (selected indices)
// ---------------------------------------------------------------------------

#define FPS_TPB 256
#define FPS_PPT 4   // points per thread kept in VGPRs

__global__ void fps_init_kernel(unsigned long long* __restrict__ best,
                                unsigned* __restrict__ cnt) {
    int i = blockIdx.x * blockDim.x + threadIdx.x;
    if (i < 1024) best[i] = 0ull;
    if (i == 0) *cnt = 0u;
}

__global__ void __launch_bounds__(FPS_TPB, 1)
fps_persistent_kernel(const float* __restrict__ coord,
                      unsigned long long* __restrict__ best,
                      unsigned* __restrict__ cnt,
                      int* __restrict__ sel,
                      int N, int M) {
    const int t    = threadIdx.x;
    const int G    = gridDim.x;
    const int base = blockIdx.x * (FPS_TPB * FPS_PPT);
    const int lane = t & 31;          // wave32
    const int wid  = t >> 5;

    // Register-resident point slice + running min distance.
    float px[FPS_PPT], py[FPS_PPT], pz[FPS_PPT], dmin[FPS_PPT];
    int   pidx[FPS_PPT];
#pragma unroll
    for (int j = 0; j < FPS_PPT; ++j) {
        int i = base + j * FPS_TPB + t;
        pidx[j] = i;
        bool v = (i < N);
        size_t a = (size_t)(v ? i : 0) * 3;
        px[j] = coord[a + 0];
        py[j] = coord[a + 1];
        pz[j] = coord[a + 2];
        dmin[j] = v ? 1e10f : -1.0f;   // pads can never win the argmax
    }

    __shared__ unsigned long long red[FPS_TPB / 32];
    __shared__ int s_far;

    int far = 0;  // reference starts at index 0

    for (int k = 0; k < M; ++k) {
        if (blockIdx.x == 0 && t == 0) sel[k] = far;

        // far is wave-uniform: pin it in an SGPR so the centroid loads
        // become scalar (SMEM) loads instead of 32-lane VMEM broadcasts.
        int sfar = __builtin_amdgcn_readfirstlane(far);
        size_t fa = (size_t)sfar * 3;
        float cx = coord[fa + 0], cy = coord[fa + 1], cz = coord[fa + 2];

        // Update running min distance; track local best (max) with
        // first-index tie-break semantics.
        float lbest = -1.0f;
        int   lidx  = 0;
#pragma unroll
        for (int j = 0; j < FPS_PPT; ++j) {
            float dx = px[j] - cx, dy = py[j] - cy, dz = pz[j] - cz;
            float d  = dx * dx + dy * dy + dz * dz;
            float dm = fminf(dmin[j], d);
            dmin[j]  = dm;
            if (dm > lbest) { lbest = dm; lidx = pidx[j]; }
        }

        // Pack: dist bits (non-negative float -> uint order) high, ~index low
        // so u64-max picks larger distance, then SMALLER index on ties
        // (matches jnp.argmax first-index semantics).
        unsigned long long packed =
            (lbest < 0.0f)
                ? 0ull
                : ((unsigned long long)__float_as_uint(lbest) << 32) |
                      (unsigned long long)(0xFFFFFFFFu - (unsigned)lidx);

        // Wave32 reduction.
#pragma unroll
        for (int off = 16; off > 0; off >>= 1) {
            unsigned long long o = __shfl_xor(packed, off, 32);
            if (o > packed) packed = o;
        }
        if (lane == 0) red[wid] = packed;
        __syncthreads();

        if (t == 0) {
            unsigned long long m = red[0];
#pragma unroll
            for (int w = 1; w < FPS_TPB / 32; ++w)
                if (red[w] > m) m = red[w];
            atomicMax(&best[k], m);
            __threadfence();
            // Grid barrier: monotonic counter, target = (k+1)*G.
            atomicAdd(cnt, 1u);
            unsigned target = (unsigned)(k + 1) * (unsigned)G;
            while (__hip_atomic_load(cnt, __ATOMIC_ACQUIRE,
                                     __HIP_MEMORY_SCOPE_AGENT) < target) {
                __builtin_amdgcn_s_sleep(2);
            }
            unsigned long long w = __hip_atomic_load(
                &best[k], __ATOMIC_RELAXED, __HIP_MEMORY_SCOPE_AGENT);
            s_far = (int)(0xFFFFFFFFu - (unsigned)w);
        }
        __syncthreads();
        far = s_far;
    }
}

// ---------------------------------------------------------------------------
// Gather: feat[idx] -> out[0 .. M*D), coord[idx] -> out[M*D .. M*D + M*3).
// Uses gfx1250 async global<->LDS B128 path (ASYNCcnt) for the row copy.
// ---------------------------------------------------------------------------
__global__ void __launch_bounds__(128)
gather_rows_kernel(const float* __restrict__ feat,
                   const float* __restrict__ coord,
                   const int* __restrict__ sel,
                   float* __restrict__ out,
                   int D, int M) {
    const int r   = blockIdx.x;
    const int t   = threadIdx.x;
    const int idx = sel[r];

    const float* src = feat + (size_t)idx * D;
    float*       dst = out + (size_t)r * D;

    __shared__ float buf[512];  // 128 threads * float4

    const int tile = (int)blockDim.x * 4;  // 512 floats per pass
    int base = 0;
    for (; base + tile <= D; base += tile) {
        unsigned lds = (unsigned)(uintptr_t)&buf[t * 4];  // LDS byte offset
        const float* g = src + base + t * 4;
        // async global -> LDS (tracked by ASYNCcnt)
        asm volatile("global_load_async_to_lds_b128 %0, %1, off"
                     :: "v"(lds), "v"(g) : "memory");
        asm volatile("s_wait_asynccnt 0x0" ::: "memory");
        __syncthreads();
        float* o = dst + base + t * 4;
        // async LDS -> global
        asm volatile("global_store_async_from_lds_b128 %0, %1, off"
                     :: "v"(o), "v"(lds) : "memory");
        // Make LDS safe to overwrite on the next tile (S_ENDPGM's implicit
        // wait-idle covers the final tile's store).
        asm volatile("s_wait_asynccnt 0x0" ::: "memory");
        __syncthreads();
    }
    // Remainder (D not a multiple of 512): plain copy.
    for (int i = base + t; i < D; i += blockDim.x) dst[i] = src[i];

    // coord gather (3 floats per row).
    if (t < 3)
        out[(size_t)M * D + (size_t)r * 3 + t] = coord[(size_t)idx * 3 + t];
}

extern "C" void kernel_launch(void* const* d_in, const int* in_sizes, int n_in,
                              void* d_out, int out_size, void* d_ws, size_t ws_size,
                              hipStream_t stream) {
    const float* feat  = (const float*)d_in[0];
    const float* coord = (const float*)d_in[1];
    float* out = (float*)d_out;

    const int N = in_sizes[1] / 3;
    const int D = (N > 0) ? (in_sizes[0] / N) : 0;
    const int M = (N < 1024) ? N : 1024;

    char* ws = (char*)d_ws;
    unsigned long long* best = (unsigned long long*)ws;          // 8192 B
    unsigned*           cnt  = (unsigned*)(ws + 8192);           // 4 B
    int*                sel  = (int*)(ws + 8448);                // 4096 B

    fps_init_kernel<<<4, 256, 0, stream>>>(best, cnt);

    const int ptsPerWG = FPS_TPB * FPS_PPT;                      // 1024
    const int G = (N + ptsPerWG - 1) / ptsPerWG;                 // 98 for N=100000
    fps_persistent_kernel<<<G, FPS_TPB, 0, stream>>>(coord, best, cnt, sel, N, M);

    gather_rows_kernel<<<M, 128, 0, stream>>>(feat, coord, sel, out, D, M);
}